// PrimaryCaps_89610197664522
// MI455X (gfx1250) — compile-verified
//
#include <hip/hip_runtime.h>

typedef float v2f __attribute__((ext_vector_type(2)));
typedef float v8f __attribute__((ext_vector_type(8)));

#define BATCH   100
#define IN_CH   256
#define KSZ     9
#define KK      81          // KSZ*KSZ
#define KPAD    84          // padded to multiple of 4 (21 WMMA steps of K=4)
#define OC_TOT  256         // NUM_CAPS * OUT_CH
#define IMG     23
#define NPOS    64          // 8x8 output positions per image
#define MT      64          // output channels per block
#define NT      64          // output positions per block (whole image)
#define KROW    (IN_CH * KK)              // 20736 floats per weight row
#define BSTRIDE (NT * KPAD)               // one B buffer: 5376 floats
#define U_ELEMS (BATCH * OC_TOT * NPOS)   // 1,638,400

// squash constants
#define T1C (-13.46416092f)
#define T3C (13.23405266f)
#define A1C (0.000242759f)
#define B1C (0.024488359f)
#define A2C (0.002769205f)
#define B2C (0.06089699f)
#define A3C (-0.002828244f)
#define B3C (0.061313814f)
#define A4C (-0.000219038f)
#define B4C (0.023874787f)

// gfx1250 async global->LDS path (ASYNCcnt-tracked), with safe fallback.
#if defined(__gfx1250__) &&                                             \
    __has_builtin(__builtin_amdgcn_global_load_async_to_lds_b32) &&     \
    __has_builtin(__builtin_amdgcn_s_wait_asynccnt)
#define USE_ASYNC_LDS 1
#define ASYNC_CP32(gptr, lptr)                                          \
  __builtin_amdgcn_global_load_async_to_lds_b32((int*)(gptr),           \
                                                (int*)(lptr), 0, 0)
#else
#define USE_ASYNC_LDS 0
#endif

// Stage one input channel's im2col row for this thread's (row, q) role:
// column lattice c = q + 4j with incremental kh/kw (no div/mod).
// Exactly 21 async issues per wave per call (group size for ASYNCcnt waits).
__device__ __forceinline__ void stage_B(const float* __restrict__ gB,
                                        float* __restrict__ lB, int q) {
  int kw  = q;                    // c = q  ->  kh=0, kw=q
  int off = q;                    // kh*IMG + kw
#pragma unroll
  for (int j = 0; j < 21; ++j) {
    const int c = q + 4 * j;
    if (c < KK) {
#if USE_ASYNC_LDS
      ASYNC_CP32(gB + off, lB + c);
#else
      lB[c] = gB[off];
#endif
    }
    kw += 4;
    off += 4;
    if (kw >= KSZ) { kw -= KSZ; off += IMG - KSZ; }
  }
}

// ---------------------------------------------------------------------------
// Conv (stride-2, VALID) as implicit GEMM on the fp32 WMMA pipe.
// Block = 256 threads = 8 wave32s; tile = 64 oc x 64 positions (one image).
// Each wave owns TWO 16x16 C tiles sharing one A fragment (4 mSub x 2 nPair).
//  - B (im2col) is DOUBLE-BUFFERED in LDS via GLOBAL_LOAD_ASYNC_TO_LDS_B32:
//    chunk ic+1 is issued before computing chunk ic; s_wait_asynccnt(21)
//    retires exactly the older group (async loads complete in order).
//  - A (weights) streams straight to VGPRs: W (21 MB) is L2-resident (192 MB)
//    and rows are contiguous in k, so LDS is not needed for A.
// K walks one input channel at a time: 81 values padded to 84 so every step
// is a clean V_WMMA_F32_16X16X4_F32 (pad handled as zero-A in registers).
// ---------------------------------------------------------------------------
__global__ __launch_bounds__(256)
void conv_wmma_kernel(const float* __restrict__ x0,
                      const float* __restrict__ Wt,
                      const float* __restrict__ bias,
                      float* __restrict__ u) {
  __shared__ float Blds[2 * BSTRIDE];   // double-buffered 64 x 84 im2col

  const int t      = threadIdx.x;
  const int lane   = t & 31;
  const int wave   = t >> 5;
  const int mSub   = wave >> 1;       // 0..3 : 16-row band of the 64 oc
  const int nPair  = wave & 1;        // 0..1 : 32-col band (two 16-col tiles)
  const int ocBase = blockIdx.x * MT;
  const int img    = blockIdx.y;

  const float* xImg = x0 + (size_t)img * IN_CH * IMG * IMG;

  // Zero pad columns (81..83) of both buffers once; staging never writes them.
  if (t < NT) {
#pragma unroll
    for (int c = KK; c < KPAD; ++c) {
      Blds[t * KPAD + c]           = 0.f;
      Blds[BSTRIDE + t * KPAD + c] = 0.f;
    }
  }

  // staging role: thread owns row (t>>2), column lattice q + 4j, j = 0..20
  const int ar = t >> 2;              // 0..63 : output position
  const int q  = t & 3;
  const int oh = ar >> 3;
  const int ow = ar & 7;
  const float* gB0 = xImg + (oh * 2) * IMG + ow * 2;
  float* lB = Blds + ar * KPAD;

  v8f acc0 = {};
  v8f acc1 = {};

  // per-wave fragment addressing (ISA fp32 A/B layouts):
  //   lane L: row/col = L%16, K = 4s + 2*(L/16) + {0,1} (contiguous pair)
  const int half2  = (lane >> 4) << 1;
  const int lane15 = lane & 15;
  const float* aRow  = Wt + (size_t)(ocBase + mSub * 16 + lane15) * KROW + half2;
  const float* Bfrag = Blds + (nPair * 32 + lane15) * KPAD + half2;

  // prologue: kick off chunk 0 into buffer 0
  stage_B(gB0, lB, q);

  for (int ic = 0; ic < IN_CH; ++ic) {
    if (ic + 1 < IN_CH) {
      // issue next chunk into the other buffer, then retire the older group
      stage_B(gB0 + (ic + 1) * (IMG * IMG), lB + ((ic + 1) & 1) * BSTRIDE, q);
#if USE_ASYNC_LDS
      __builtin_amdgcn_s_wait_asynccnt(21);
#endif
    } else {
#if USE_ASYNC_LDS
      __builtin_amdgcn_s_wait_asynccnt(0);
#endif
    }
    __syncthreads();

    const float* aBase = aRow + ic * KK;           // A from global (L2-hot)
    const float* Bc0   = Bfrag + (ic & 1) * BSTRIDE;
    const float* Bc1   = Bc0 + 16 * KPAD;

    // ---- 20 full WMMA steps x 2 N-tiles ----------------------------------
#pragma unroll
    for (int s = 0; s < 20; ++s) {
      v2f a;
      a.x = aBase[4 * s];
      a.y = aBase[4 * s + 1];
      v2f b0 = *(const v2f*)(Bc0 + 4 * s);
      v2f b1 = *(const v2f*)(Bc1 + 4 * s);
      acc0 = __builtin_amdgcn_wmma_f32_16x16x4_f32(
          false, a, false, b0, (short)0, acc0, false, false);
      acc1 = __builtin_amdgcn_wmma_f32_16x16x4_f32(
          false, a, false, b1, (short)0, acc1, false, false);
    }
    // ---- step s=20 covers k=80..83: only k=80 is real (lanes 0-15) -------
    {
      v2f a;
      a.x = (half2 == 0) ? aBase[80] : 0.f;
      a.y = 0.f;
      v2f b0 = *(const v2f*)(Bc0 + 80);
      v2f b1 = *(const v2f*)(Bc1 + 80);
      acc0 = __builtin_amdgcn_wmma_f32_16x16x4_f32(
          false, a, false, b0, (short)0, acc0, false, false);
      acc1 = __builtin_amdgcn_wmma_f32_16x16x4_f32(
          false, a, false, b1, (short)0, acc1, false, false);
    }
    __syncthreads();   // everyone done reading buf[ic&1] before it's refilled
  }

  // ---- epilogue: C layout -> u[img][oc*64 + pos], fused bias -------------
  // VGPR r: lanes 0-15 hold M=r, lanes 16-31 hold M=r+8; N = lane%16.
  const int mOff  = (lane >> 4) * 8;
  const int ocRow = ocBase + mSub * 16 + mOff;
  const int lp0   = nPair * 32 + lane15;
  const int lp1   = lp0 + 16;
  float* uImg = u + (size_t)img * (OC_TOT * NPOS);
#pragma unroll
  for (int r = 0; r < 8; ++r) {
    const int oc = ocRow + r;
    const float bv = bias[oc];
    uImg[oc * NPOS + lp0] = acc0[r] + bv;
    uImg[oc * NPOS + lp1] = acc1[r] + bv;
  }
}

// ---------------------------------------------------------------------------
// Rank-based squash: no sort needed. Each element's fate depends only on its
// rank (stable, tie-broken by index == jnp.argsort semantics) and the scalar
// counts i1/i2/i3, which are workgroup reductions. One block per batch.
// ---------------------------------------------------------------------------
__global__ __launch_bounds__(256)
void squash_kernel(const float* __restrict__ u, float* __restrict__ sq) {
  __shared__ float f[2048];
  __shared__ int cnt;

  const int t = threadIdx.x;
  const int b = blockIdx.x;
  const float* ub = u + (size_t)b * (OC_TOT * NPOS);

  for (int r = t; r < 2048; r += 256) f[r] = ub[r * 8];  // first = u[:,:,0]
  __syncthreads();

  int   rank[8];
  float val[8];
  float cur[8];
#pragma unroll
  for (int i = 0; i < 8; ++i) {
    const int e = t + 256 * i;
    const float v = f[e];
    int rk = 0;
    for (int m = 0; m < 2048; m += 4) {
      const float4 fm = *(const float4*)(f + m);
      rk += (fm.x < v) || (fm.x == v && (m + 0) < e);
      rk += (fm.y < v) || (fm.y == v && (m + 1) < e);
      rk += (fm.z < v) || (fm.z == v && (m + 2) < e);
      rk += (fm.w < v) || (fm.w == v && (m + 3) < e);
    }
    rank[i] = rk;
    val[i]  = v;
  }

  // ---- i1 = count(first < T1) --------------------------------------------
  if (t == 0) cnt = 0;
  __syncthreads();
  int loc = 0;
#pragma unroll
  for (int i = 0; i < 8; ++i) loc += (val[i] < T1C) ? 1 : 0;
  atomicAdd(&cnt, loc);
  __syncthreads();
  const int i1 = cnt;
  __syncthreads();

#pragma unroll
  for (int i = 0; i < 8; ++i)
    cur[i] = (rank[i] < i1 - 1) ? (A1C * val[i] + B1C) : val[i];

  // ---- i2 = count(cur < 0) -----------------------------------------------
  if (t == 0) cnt = 0;
  __syncthreads();
  loc = 0;
#pragma unroll
  for (int i = 0; i < 8; ++i) loc += (cur[i] < 0.f) ? 1 : 0;
  atomicAdd(&cnt, loc);
  __syncthreads();
  const int i2 = cnt;
  __syncthreads();

  const bool c2 = (i2 > 0) && (i2 > i1);
#pragma unroll
  for (int i = 0; i < 8; ++i)
    if (c2 && rank[i] >= i1 && rank[i] < i2 - 1) cur[i] = A2C * val[i] + B2C;

  // ---- i3 = count(cur < T3) ----------------------------------------------
  if (t == 0) cnt = 0;
  __syncthreads();
  loc = 0;
#pragma unroll
  for (int i = 0; i < 8; ++i) loc += (cur[i] < T3C) ? 1 : 0;
  atomicAdd(&cnt, loc);
  __syncthreads();
  const int i3 = cnt;

  const bool c3 = (i3 > 0) && (i3 > i2);
#pragma unroll
  for (int i = 0; i < 8; ++i) {
    if (c3 && rank[i] >= i2 && rank[i] < i3 - 1) cur[i] = A3C * val[i] + B3C;
    if ((i3 < 2048) && rank[i] >= i3 && rank[i] < 2047)
      cur[i] = A4C * val[i] + B4C;
  }

  // ---- squashed = mag[:,None] * u ----------------------------------------
  float* sqb = sq + (size_t)b * (OC_TOT * NPOS);
#pragma unroll
  for (int i = 0; i < 8; ++i) {
    const int e = t + 256 * i;
    const float m = cur[i];
    const float4 u0 = *(const float4*)(ub + e * 8);
    const float4 u1 = *(const float4*)(ub + e * 8 + 4);
    float4 s0 = make_float4(m * u0.x, m * u0.y, m * u0.z, m * u0.w);
    float4 s1 = make_float4(m * u1.x, m * u1.y, m * u1.z, m * u1.w);
    *(float4*)(sqb + e * 8)     = s0;
    *(float4*)(sqb + e * 8 + 4) = s1;
  }
}

__global__ void copy_labels_kernel(const int* __restrict__ lbl, int* __restrict__ dst) {
  const int i = blockIdx.x * blockDim.x + threadIdx.x;
  if (i < BATCH) dst[i] = lbl[i];
}

extern "C" void kernel_launch(void* const* d_in, const int* in_sizes, int n_in,
                              void* d_out, int out_size, void* d_ws, size_t ws_size,
                              hipStream_t stream) {
  const float* x0    = (const float*)d_in[0];
  const int*   mnist = (const int*)d_in[1];
  // d_in[2]=train_or_test, d_in[3]=epch (unused)
  const float* Wt    = (const float*)d_in[4];
  const float* bias  = (const float*)d_in[5];

  float* out     = (float*)d_out;
  float* sq_out  = out;                 // squashed: U_ELEMS floats
  float* u_out   = out + U_ELEMS;       // u:        U_ELEMS floats
  int*   lbl_out = (int*)(out + 2 * (size_t)U_ELEMS);

  dim3 gridConv(OC_TOT / MT, BATCH);    // (4, 100)
  conv_wmma_kernel<<<gridConv, 256, 0, stream>>>(x0, Wt, bias, u_out);

  squash_kernel<<<dim3(BATCH), 256, 0, stream>>>(u_out, sq_out);

  copy_labels_kernel<<<dim3(1), 128, 0, stream>>>(mnist, lbl_out);
}